// RPN_27736898798121
// MI455X (gfx1250) — compile-verified
//
#include <hip/hip_runtime.h>
#include <hip/hip_bf16.h>

// ---------------- problem constants (match reference) ----------------
constexpr int   kB    = 16;
constexpr int   kN    = 300000;
constexpr int   kPRE  = 2000;
constexpr int   kPREP = 2048;   // padded rows (multiple of 256)
constexpr int   kCAP  = 4096;   // candidate cap before exact sort
constexpr int   kPOST = 1000;
constexpr float kThr  = 0.7f;
constexpr float kMinSize = 0.001f;
constexpr float kImgW = 1333.0f;
constexpr float kImgH = 800.0f;
constexpr int   kWORDS = kPREP / 32;   // 64 keep/sup words per row

__device__ __forceinline__ unsigned sortkey(float x) {
    unsigned u = __float_as_uint(x);
    // monotone map: larger float -> larger unsigned
    u ^= (unsigned)((int)u >> 31) | 0x80000000u;
    return u;
}

// ============================================================================
// K1: per-image 2-level histogram radix-select of top-PRE scores, compaction
// ============================================================================
__global__ void __launch_bounds__(1024)
k_select(const float* __restrict__ cls,
         unsigned long long* __restrict__ cand,   // [B][kCAP]
         unsigned* __restrict__ candCnt)          // [B]
{
    __shared__ unsigned hist[1024];
    __shared__ unsigned sT1, sT2, sAbove, sCnt;

    const int img = blockIdx.x;
    const int tid = threadIdx.x;
    const float* sc = cls + (size_t)img * kN;

    // ---- pass 1: coarse histogram on key[31:22] ----
    for (int b = tid; b < 1024; b += 1024) hist[b] = 0u;
    if (tid == 0) sCnt = 0u;
    __syncthreads();
    for (int n = tid; n < kN; n += 1024) {
        __builtin_prefetch(sc + n + 4096, 0, 0);       // global_prefetch_b8
        atomicAdd(&hist[sortkey(sc[n]) >> 22], 1u);
    }
    __syncthreads();
    if (tid == 0) {
        unsigned acc = 0; int b;
        sT1 = 0; sAbove = 0;
        for (b = 1023; b >= 0; --b) {
            unsigned h = hist[b];
            if (acc + h >= (unsigned)kPRE) { sT1 = (unsigned)b; sAbove = acc; break; }
            acc += h;
        }
        if (b < 0) { sT1 = 0u; sAbove = acc; }
    }
    __syncthreads();
    const unsigned T1 = sT1, above = sAbove;

    // ---- pass 2: refine inside coarse bin T1 on key[21:12] ----
    for (int b = tid; b < 1024; b += 1024) hist[b] = 0u;
    __syncthreads();
    for (int n = tid; n < kN; n += 1024) {
        unsigned u = sortkey(sc[n]);
        if ((u >> 22) == T1) atomicAdd(&hist[(u >> 12) & 1023u], 1u);
    }
    __syncthreads();
    if (tid == 0) {
        unsigned acc = above; int b;
        sT2 = 0;
        for (b = 1023; b >= 0; --b) {
            acc += hist[b];
            if (acc >= (unsigned)kPRE) { sT2 = (unsigned)b; break; }
        }
        if (b < 0) sT2 = 0u;
    }
    __syncthreads();
    const unsigned T2 = sT2;

    // ---- pass 3: compact candidates ----
    for (int n = tid; n < kN; n += 1024) {
        unsigned u = sortkey(sc[n]);
        unsigned b = u >> 22;
        if (b > T1 || (b == T1 && ((u >> 12) & 1023u) >= T2)) {
            unsigned p = atomicAdd(&sCnt, 1u);
            if (p < (unsigned)kCAP) {
                // tie-break: equal score -> smaller index first (matches top_k)
                unsigned long long key =
                    ((unsigned long long)u << 32) | (0xFFFFFFFFu - (unsigned)n);
                cand[(size_t)img * kCAP + p] = key;
            }
        }
    }
    __syncthreads();
    if (tid == 0) candCnt[img] = sCnt < (unsigned)kCAP ? sCnt : (unsigned)kCAP;
}

// ============================================================================
// K2: bitonic sort candidates (desc), decode/clip top-PRE, stable partition
//     valid-first, write padded box rows + numValid
// ============================================================================
__global__ void __launch_bounds__(1024)
k_sort_decode(const float* __restrict__ reg,
              const float* __restrict__ anchors,
              const unsigned long long* __restrict__ cand,
              const unsigned* __restrict__ candCnt,
              float* __restrict__ boxesOut,        // [B][kPREP][4]
              unsigned* __restrict__ numValid)     // [B]
{
    __shared__ unsigned long long key[kCAP];       // 32 KB
    __shared__ unsigned vflag[kPREP];              // 8 KB
    __shared__ unsigned wword[kWORDS];
    __shared__ unsigned wpref[kWORDS + 1];

    const int img = blockIdx.x;
    const int tid = threadIdx.x;

    const unsigned cnt0 = candCnt[img];
    const unsigned cnt  = cnt0 < (unsigned)kCAP ? cnt0 : (unsigned)kCAP;
    for (int i = tid; i < kCAP; i += 1024)
        key[i] = (i < (int)cnt) ? cand[(size_t)img * kCAP + i] : 0ull;
    __syncthreads();

    // bitonic sort, descending
    for (unsigned k = 2; k <= (unsigned)kCAP; k <<= 1) {
        for (unsigned j = k >> 1; j > 0; j >>= 1) {
            for (unsigned i = tid; i < (unsigned)kCAP; i += 1024) {
                unsigned ixj = i ^ j;
                if (ixj > i) {
                    unsigned long long a = key[i], b = key[ixj];
                    bool descBlk = ((i & k) == 0);
                    if (descBlk ? (a < b) : (a > b)) { key[i] = b; key[ixj] = a; }
                }
            }
            __syncthreads();
        }
    }

    // decode top-PRE (2 elements per thread)
    float bx[2][4];
    int   gi[2]   = {-1, -1};
    for (int e = 0; e < 2; ++e) {
        int i = tid + e * 1024;
        if (i < kPRE) {
            gi[e] = i;
            unsigned long long kk = key[i];
            unsigned idx = 0xFFFFFFFFu - (unsigned)(kk & 0xFFFFFFFFull);
            bool ok = (kk >> 32) != 0ull && idx < (unsigned)kN;
            float x1 = 0.f, y1 = 0.f, x2 = 0.f, y2 = 0.f;
            bool valid = false;
            if (ok) {
                const float* a = anchors + ((size_t)img * kN + idx) * 4;
                const float* r = reg     + ((size_t)img * kN + idx) * 4;
                float aw = a[2] - a[0], ah = a[3] - a[1];
                float acx = a[0] + 0.5f * aw, acy = a[1] + 0.5f * ah;
                float pcx = r[0] * aw + acx;
                float pcy = r[1] * ah + acy;
                float pw = expf(r[2]) * aw;
                float ph = expf(r[3]) * ah;
                x1 = fminf(fmaxf(pcx - 0.5f * pw, 0.f), kImgW);
                y1 = fminf(fmaxf(pcy - 0.5f * ph, 0.f), kImgH);
                x2 = fminf(fmaxf(pcx + 0.5f * pw, 0.f), kImgW);
                y2 = fminf(fmaxf(pcy + 0.5f * ph, 0.f), kImgH);
                valid = ((x2 - x1) >= kMinSize) && ((y2 - y1) >= kMinSize);
            }
            bx[e][0] = x1; bx[e][1] = y1; bx[e][2] = x2; bx[e][3] = y2;
            vflag[i] = valid ? 1u : 0u;
        }
    }
    for (int i = kPRE + tid; i < kPREP; i += 1024) vflag[i] = 0u;
    __syncthreads();

    // pack valid bits, exclusive scan over words
    if (tid < kWORDS) {
        unsigned w = 0;
        for (int b = 0; b < 32; ++b) w |= (vflag[tid * 32 + b] & 1u) << b;
        wword[tid] = w;
    }
    __syncthreads();
    if (tid == 0) {
        unsigned acc = 0;
        for (int w = 0; w < kWORDS; ++w) { wpref[w] = acc; acc += __popc(wword[w]); }
        wpref[kWORDS] = acc;
    }
    __syncthreads();
    const unsigned nv = wpref[kWORDS];
    if (tid == 0) numValid[img] = nv;

    // stable partition: valid entries first (score order), then invalid
    for (int e = 0; e < 2; ++e) {
        if (gi[e] >= 0) {
            int i = gi[e];
            unsigned p = wpref[i >> 5] + __popc(wword[i >> 5] & ((1u << (i & 31)) - 1u));
            unsigned pos = vflag[i] ? p : nv + (unsigned)i - p;
            float* o = boxesOut + ((size_t)img * kPREP + pos) * 4;
            o[0] = bx[e][0]; o[1] = bx[e][1]; o[2] = bx[e][2]; o[3] = bx[e][3];
        }
    }
    for (int i = kPRE + tid; i < kPREP; i += 1024) {
        float* o = boxesOut + ((size_t)img * kPREP + i) * 4;
        o[0] = 0.f; o[1] = 0.f; o[2] = 0.f; o[3] = 0.f;
    }
}

// ============================================================================
// K3: suppression bit matrix, ROW-major: sup[r][w] (bit set iff j>i, IoU>thr).
//     Column-box tiles staged into LDS with CDNA5 async global->LDS copies.
//     Per-thread rows accumulate 4 words -> 16B uint4 stores.
// ============================================================================
__global__ void __launch_bounds__(256)
k_sup(const float* __restrict__ boxes,            // [B][kPREP][4]
      unsigned* __restrict__ sup)                 // [B][kPREP][kWORDS]
{
    __shared__ float4 tile[256];                  // LDS offset 0 (only LDS var)

    const int img = blockIdx.x;
    const int tid = threadIdx.x;
    const int r   = blockIdx.y * 256 + tid;       // row 0..kPREP-1

    const float* bp = boxes + ((size_t)img * kPREP + r) * 4;
    const float rx1 = bp[0], ry1 = bp[1], rx2 = bp[2], ry2 = bp[3];
    const float rarea = (rx2 - rx1) * (ry2 - ry1);

    unsigned* supRow = sup + ((size_t)img * kPREP + r) * kWORDS;  // 256B-aligned

    for (int t = 0; t < kPREP / 256; ++t) {
        __syncthreads();   // previous tile fully consumed
        // async stage: each lane copies one 16B box into LDS (ASYNCcnt path)
        {
            unsigned lds_off = (unsigned)(tid * 16);
            const float* gsrc = boxes + ((size_t)img * kPREP + t * 256 + tid) * 4;
            asm volatile("global_load_async_to_lds_b128 %0, %1, off"
                         :: "v"(lds_off), "v"(gsrc) : "memory");
            asm volatile("s_wait_asynccnt 0" ::: "memory");
        }
        __syncthreads();   // LDS tile visible to all waves

        for (int half = 0; half < 2; ++half) {
            unsigned mw[4];
            for (int w = 0; w < 4; ++w) {
                unsigned m = 0;
                const int jbase = t * 256 + half * 128 + w * 32;
                for (int b = 0; b < 32; ++b) {
                    float4 q = tile[half * 128 + w * 32 + b];
                    float ix1 = fmaxf(rx1, q.x), iy1 = fmaxf(ry1, q.y);
                    float ix2 = fminf(rx2, q.z), iy2 = fminf(ry2, q.w);
                    float iw = fmaxf(ix2 - ix1, 0.f);
                    float ih = fmaxf(iy2 - iy1, 0.f);
                    float inter = iw * ih;
                    float qarea = (q.z - q.x) * (q.w - q.y);
                    float uni   = fmaxf(rarea + qarea - inter, 1e-8f);
                    bool s = ((inter / uni) > kThr) && ((jbase + b) > r);
                    m |= (unsigned)s << b;
                }
                mw[w] = m;
            }
            uint4 v; v.x = mw[0]; v.y = mw[1]; v.z = mw[2]; v.w = mw[3];
            *(uint4*)(supRow + t * 8 + half * 4) = v;      // 16B-aligned store
        }
    }
}

// ============================================================================
// K4: single-wave greedy NMS (wave32, barrier-free serial chain) + emit
//     first POST kept boxes. Each lane owns 2 adjacent keep words (uint2);
//     decision bit broadcast via __shfl; sup rows are contiguous 256B reads,
//     pipelined 1 row ahead + prefetched 8 rows ahead.
// ============================================================================
__global__ void __launch_bounds__(32)
k_nms_out(const float* __restrict__ boxes,        // [B][kPREP][4]
          const unsigned* __restrict__ sup,       // [B][kPREP][kWORDS]
          const unsigned* __restrict__ numValid,  // [B]
          float* __restrict__ out)                // [B][kPOST][4]
{
    __shared__ unsigned keepw[kWORDS];
    __shared__ unsigned wpref[kWORDS + 1];

    const int img  = blockIdx.x;
    const int lane = threadIdx.x;                 // 0..31, one wave

    float* oimg = out + (size_t)img * kPOST * 4;
    for (int i = lane; i < kPOST * 4; i += 32) oimg[i] = 0.f;

    const unsigned nv = numValid[img];
    auto initw = [&](int w) -> unsigned {
        unsigned lo = (unsigned)w * 32u;
        return (nv >= lo + 32u) ? 0xFFFFFFFFu
             : (nv > lo ? ((1u << (nv - lo)) - 1u) : 0u);
    };
    unsigned keepA = initw(2 * lane);             // word 2*lane
    unsigned keepB = initw(2 * lane + 1);         // word 2*lane+1

    const uint2* supImg = (const uint2*)(sup + (size_t)img * kPREP * kWORDS);
    // row i = supImg[i*32 + lane]  (contiguous 256B per row, coalesced)

    uint2 cur = supImg[lane];                     // preload row 0
    for (int i = 0; i < kPRE; ++i) {
        uint2 c = cur;
        cur = supImg[(size_t)(i + 1) * 32 + lane];            // rows < kPREP: in-range
        __builtin_prefetch(supImg + (size_t)(i + 8) * 32 + lane, 0, 0);
        const int wi = i >> 5;                    // uniform
        unsigned sel = (wi & 1) ? keepB : keepA;
        unsigned wv  = __shfl(sel, wi >> 1, 32);  // broadcast owning word
        if ((wv >> (i & 31)) & 1u) {              // row i still alive?
            keepA &= ~c.x;
            keepB &= ~c.y;
        }
    }

    keepw[2 * lane]     = keepA;
    keepw[2 * lane + 1] = keepB;
    __syncthreads();
    if (lane == 0) {
        unsigned acc = 0;
        for (int w = 0; w < kWORDS; ++w) { wpref[w] = acc; acc += __popc(keepw[w]); }
        wpref[kWORDS] = acc;
    }
    __syncthreads();

    for (int i = lane; i < kPRE; i += 32) {
        unsigned wv = keepw[i >> 5];
        if ((wv >> (i & 31)) & 1u) {
            unsigned p = wpref[i >> 5] + __popc(wv & ((1u << (i & 31)) - 1u));
            if (p < (unsigned)kPOST) {
                const float4 v = *(const float4*)(boxes + ((size_t)img * kPREP + i) * 4);
                *(float4*)(oimg + p * 4) = v;
            }
        }
    }
}

// ============================================================================
// launcher
// ============================================================================
extern "C" void kernel_launch(void* const* d_in, const int* in_sizes, int n_in,
                              void* d_out, int out_size, void* d_ws, size_t ws_size,
                              hipStream_t stream) {
    const float* cls     = (const float*)d_in[0];   // (B, N)
    const float* reg     = (const float*)d_in[1];   // (B, N, 4)
    const float* anchors = (const float*)d_in[2];   // (B, N, 4)
    float* out = (float*)d_out;                     // (B, POST, 4)

    char* ws = (char*)d_ws;
    size_t off = 0;
    auto alloc = [&](size_t bytes) -> void* {
        void* p = ws + off;
        off = (off + bytes + 255) & ~(size_t)255;
        return p;
    };
    unsigned long long* cand   = (unsigned long long*)alloc((size_t)kB * kCAP * 8);
    unsigned*           candCnt= (unsigned*)alloc((size_t)kB * 4);
    unsigned*           numVal = (unsigned*)alloc((size_t)kB * 4);
    float*              boxesWS= (float*)alloc((size_t)kB * kPREP * 4 * 4);
    unsigned*           sup    = (unsigned*)alloc((size_t)kB * kPREP * kWORDS * 4);

    k_select     <<<kB, 1024, 0, stream>>>(cls, cand, candCnt);
    k_sort_decode<<<kB, 1024, 0, stream>>>(reg, anchors, cand, candCnt, boxesWS, numVal);
    k_sup        <<<dim3(kB, kPREP / 256), 256, 0, stream>>>(boxesWS, sup);
    k_nms_out    <<<kB, 32, 0, stream>>>(boxesWS, sup, numVal, out);
}